// Net_21509196218620
// MI455X (gfx1250) — compile-verified
//
#include <hip/hip_runtime.h>

// Rx(theta) on qubit 0 of a 24-qubit state (bandwidth-bound: ~256 MB traffic,
// ~0.13 GFLOP -> AI ~0.5 flop/byte; floor ~11 us at 23.3 TB/s).
// Real 4x4 form on v = (re0, im0, re1, im1):
//   R = [[ cs,  0,   0,  sn],
//        [  0, cs, -sn,   0],
//        [  0, sn,  cs,   0],
//        [-sn,  0,   0,  cs]]    cs=cos(t/2), sn=sin(t/2)
// Routed through V_WMMA_F32_16X16X4_F32 as D = A(16x4)*B(4x16) + C with
// B = data (4 components x 16 sites), A = R:
//   WMMA #1: R in A-rows 0-3  -> outputs land in lanes 0-15 of acc VGPRs 0-3
//   WMMA #2: R in A-rows 8-11 -> outputs land in lanes 16-31 (same accumulator)
// so acc[0..3] hold {out0_re,out0_im,out1_re,out1_im} for 32 consecutive sites
// across all 32 lanes: fully coalesced full-wave NT loads and stores.

typedef float v2f __attribute__((ext_vector_type(2)));
typedef float v8f __attribute__((ext_vector_type(8)));

__device__ __forceinline__ float sel(bool c, float v) { return c ? v : 0.f; }

__global__ void __launch_bounds__(256)
rx_gate_wmma(const float* __restrict__ re,
             const float* __restrict__ im,
             const float* __restrict__ theta,
             float* __restrict__ out,
             int H, int numPairs, int strideWaves)
{
    const int lane = threadIdx.x & 31;
    const int n    = lane & 15;   // output column / A-row selector
    const int half = lane >> 4;   // 0: K=0,1 (component 0)  1: K=2,3 (component 1)

    // Wave-uniform loop index -> scalar (SALU) loop control, EXEC stays all-1s.
    int wid = (int)(blockIdx.x * (blockDim.x >> 5) + (threadIdx.x >> 5));
    wid = __builtin_amdgcn_readfirstlane(wid);

    const float t2 = theta[0] * 0.5f;
    const float cs = cosf(t2);
    const float sn = sinf(t2);

    // A-operand f32 16x4 layout: lane m holds A[m,K=0](v0), A[m,K=1](v1);
    // lane m+16 holds A[m,K=2](v0), A[m,K=3](v1).  Per-row (v0,v1) of R at
    // columns (2*half, 2*half+1):
    const float e00 = half ? 0.f : cs;   const float e01 = half ?  sn : 0.f; // row 0
    const float e10 = half ? -sn : 0.f;  const float e11 = half ? 0.f : cs;  // row 1
    const float e20 = half ?  cs : 0.f;  const float e21 = half ? 0.f : sn;  // row 2
    const float e30 = half ? 0.f : -sn;  const float e31 = half ?  cs : 0.f; // row 3

    // Branchless (flat sum of single-level selects -> v_cmp + v_cndmask).
    v2f A1, A2;   // A1: R in rows 0-3;  A2: R in rows 8-11
    A1.x = sel(n==0,e00) + sel(n==1,e10) + sel(n==2, e20) + sel(n==3, e30);
    A1.y = sel(n==0,e01) + sel(n==1,e11) + sel(n==2, e21) + sel(n==3, e31);
    A2.x = sel(n==8,e00) + sel(n==9,e10) + sel(n==10,e20) + sel(n==11,e30);
    A2.y = sel(n==8,e01) + sel(n==9,e11) + sel(n==10,e21) + sel(n==11,e31);

    const int N2 = H << 1;  // full state length

    for (int p = wid; p < numPairs; p += strideWaves) {
        // One pair = 64 consecutive sites = two 32-site tiles (two acc chains).
        // B-operand f32 4x16 layout: VGPR0 = K0 (lanes 0-15) / K2 (lanes 16-31),
        // VGPR1 = K1 / K3.  K0=re0, K1=im0, K2=re1, K3=im1.
        const int s = (p << 6) + n + half * H;
        v2f B1, B2, B3, B4;
        B1.x = __builtin_nontemporal_load(re + s);
        B1.y = __builtin_nontemporal_load(im + s);
        B2.x = __builtin_nontemporal_load(re + s + 16);
        B2.y = __builtin_nontemporal_load(im + s + 16);
        B3.x = __builtin_nontemporal_load(re + s + 32);
        B3.y = __builtin_nontemporal_load(im + s + 32);
        B4.x = __builtin_nontemporal_load(re + s + 48);
        B4.y = __builtin_nontemporal_load(im + s + 48);

        v8f acc0 = {0.f,0.f,0.f,0.f,0.f,0.f,0.f,0.f};
        v8f acc1 = {0.f,0.f,0.f,0.f,0.f,0.f,0.f,0.f};
        // (neg_a, A, neg_b, B, c_mod, C, reuse_a, reuse_b)
        acc0 = __builtin_amdgcn_wmma_f32_16x16x4_f32(false, A1, false, B1,
                                                     (short)0, acc0, false, false);
        acc1 = __builtin_amdgcn_wmma_f32_16x16x4_f32(false, A1, false, B3,
                                                     (short)0, acc1, false, false);
        acc0 = __builtin_amdgcn_wmma_f32_16x16x4_f32(false, A2, false, B2,
                                                     (short)0, acc0, false, false);
        acc1 = __builtin_amdgcn_wmma_f32_16x16x4_f32(false, A2, false, B4,
                                                     (short)0, acc1, false, false);

        // C/D layout: VGPR r = row r (lanes 0-15) and row r+8 (lanes 16-31).
        // Output: out[k]=out0_re, out[H+k]=out1_re, out[N+k]=out0_im,
        //         out[N+H+k]=out1_im.
        const int ob = (p << 6) + lane;
        __builtin_nontemporal_store(acc0[0], out + ob);
        __builtin_nontemporal_store(acc0[2], out + H  + ob);
        __builtin_nontemporal_store(acc0[1], out + N2 + ob);
        __builtin_nontemporal_store(acc0[3], out + N2 + H + ob);
        __builtin_nontemporal_store(acc1[0], out + ob + 32);
        __builtin_nontemporal_store(acc1[2], out + H  + ob + 32);
        __builtin_nontemporal_store(acc1[1], out + N2 + ob + 32);
        __builtin_nontemporal_store(acc1[3], out + N2 + H + ob + 32);
    }
}

extern "C" void kernel_launch(void* const* d_in, const int* in_sizes, int n_in,
                              void* d_out, int out_size, void* d_ws, size_t ws_size,
                              hipStream_t stream) {
    const float* re    = (const float*)d_in[0];
    const float* im    = (const float*)d_in[1];
    const float* theta = (const float*)d_in[2];
    float* out = (float*)d_out;

    const int n = in_sizes[0];        // 2^24 amplitudes
    const int H = n >> 1;             // 2^23 sites
    const int numPairs = H >> 6;      // 64 sites per wave-iteration

    const int pairsPerWave  = 4;      // short per-wave loop for latency hiding
    const int threads       = 256;    // 8 wave32s per workgroup
    const int wavesPerBlock = threads / 32;
    int wavesTotal = (numPairs + pairsPerWave - 1) / pairsPerWave;
    if (wavesTotal < 1) wavesTotal = 1;
    int blocks = (wavesTotal + wavesPerBlock - 1) / wavesPerBlock;
    const int strideWaves = blocks * wavesPerBlock;

    rx_gate_wmma<<<blocks, threads, 0, stream>>>(re, im, theta, out,
                                                 H, numPairs, strideWaves);
}